// CFA_80779744903696
// MI455X (gfx1250) — compile-verified
//
#include <hip/hip_runtime.h>
#include <math.h>

// ---------------------------------------------------------------------------
// CFA score kernel for MI455X (gfx1250, wave32).
//   dist^2[r,m] = ||phi[r]||^2 + ||C[:,m]||^2 - 2*phi[r].C[:,m]
//   per-row top-3 smallest, score = softmin(d)[0]*d[0]
// GEMM core: V_WMMA_F32_16X16X4_F32 (exact fp32, fp32 accumulate).
// Staging: double-buffered GLOBAL_LOAD_ASYNC_TO_LDS (ASYNCcnt) when available.
// ---------------------------------------------------------------------------

typedef __attribute__((ext_vector_type(2))) float v2f;
typedef __attribute__((ext_vector_type(8))) float v8f;
typedef __attribute__((ext_vector_type(4))) int   v4i;

#define R_TOTAL 50176   // B*HW = 16*3136
#define KDIM    1792    // C
#define MDIM    3136    // M
#define MPAD    3200    // padded M (multiple of 128)
#define RT      128     // rows per workgroup
#define NTILE   128     // cols per n-iteration
#define NITER   25      // ceil(3136/128)
#define KT      32      // K chunk staged in LDS
#define KITER   (KDIM / KT)   // 56
#define A_STR   36      // A row stride (floats): 16B aligned, frag b64 conflict-free
#define CSTR    144     // B pair-row stride (float2): 2*CSTR%64(dw)==32 -> lane halves disjoint

// ---- async global->LDS copy path (CDNA5), guarded with sync fallback ------
#if __has_builtin(__builtin_amdgcn_global_load_async_to_lds_b128) && \
    __has_builtin(__builtin_amdgcn_global_load_async_to_lds_b32)
#define USE_ASYNC 1
// Signatures (from clang diagnostics): (AS1 typed src, AS3 typed dst, Ii, Ii)
#define ASYNC_CP_B128(g, l)                                             \
  __builtin_amdgcn_global_load_async_to_lds_b128(                       \
      (__attribute__((address_space(1))) v4i*)(uintptr_t)(g),           \
      (__attribute__((address_space(3))) v4i*)(uint32_t)(uintptr_t)(l), \
      0, 0)
#define ASYNC_CP_B32(g, l)                                              \
  __builtin_amdgcn_global_load_async_to_lds_b32(                        \
      (__attribute__((address_space(1))) int*)(uintptr_t)(g),           \
      (__attribute__((address_space(3))) int*)(uint32_t)(uintptr_t)(l), \
      0, 0)
#if __has_builtin(__builtin_amdgcn_s_wait_asynccnt)
#define WAIT_ASYNC(n) __builtin_amdgcn_s_wait_asynccnt(n)
#else
#define WAIT_ASYNC(n) asm volatile("s_wait_asynccnt %0" ::"i"(n) : "memory")
#endif
#else
#define USE_ASYNC 0
#define WAIT_ASYNC(n) ((void)0)
#endif

// ---------------- prekernels: row / column squared norms -------------------

__global__ __launch_bounds__(256) void row_norms_k(const float* __restrict__ A,
                                                   float* __restrict__ rn) {
  const int tid  = threadIdx.x;
  const int lane = tid & 31;
  const int wv   = tid >> 5;
  const int row  = blockIdx.x * 8 + wv;          // one wave32 per row
  const float* p = A + (size_t)row * KDIM;
  float s = 0.f;
  for (int c = lane; c < KDIM; c += 32) { float v = p[c]; s += v * v; }
  #pragma unroll
  for (int off = 16; off > 0; off >>= 1) s += __shfl_down(s, off, 32);
  if (lane == 0) rn[row] = s;
}

__global__ __launch_bounds__(256) void col_norms_k(const float* __restrict__ Cb,
                                                   float* __restrict__ cn) {
  const int m = blockIdx.x * 256 + threadIdx.x;
  if (m >= MPAD) return;
  if (m >= MDIM) { cn[m] = 1e30f; return; }      // pad cols -> never in top-3
  float s = 0.f;
  for (int c = 0; c < KDIM; ++c) { float v = Cb[(size_t)c * MDIM + m]; s += v * v; }
  cn[m] = s;
}

// ---------------- fused WMMA GEMM + running top-3 --------------------------

union SMem {
  struct {
    float a[2][RT * A_STR];            // double-buffered A: 2 x 18,432 B
    float b[2][(KT / 2) * CSTR * 2];   // double-buffered B (pair-interleaved): 2 x 18,432 B
  } stg;                               // 73,728 B total
  float red[RT * 16 * 3];              // final per-row merge (24,576 B, overlaid)
};

__global__ __launch_bounds__(256) void cfa_topk_k(const float* __restrict__ Ag,
                                                  const float* __restrict__ Bg,
                                                  const float* __restrict__ rn,
                                                  const float* __restrict__ cn,
                                                  float* __restrict__ out) {
  __shared__ SMem sm;
  const int tid   = threadIdx.x;
  const int wave  = tid >> 5;          // 0..7, each wave owns 16 rows
  const int lane  = tid & 31;
  const int lrow  = lane & 15;
  const int khalf = lane >> 4;         // selects K pair / M half per WMMA layout
  const size_t rbase = (size_t)blockIdx.x * RT;

  // Stage one K-chunk (A: 4 x async b128, B: 16 x async b32, pair-interleaved).
  // Always exactly 20 async ops per lane -> wave ASYNCcnt bookkeeping is exact.
  auto stage = [&](int col0_, int kt_, int buf_) {
    const int k0 = kt_ * KT;
    #pragma unroll
    for (int i = 0; i < 4; ++i) {                 // A tile: 128 rows x 32 floats
      const int idx = tid + i * 256;
      const int r   = idx >> 3;
      const int c4  = (idx & 7) << 2;
      const float* gp = Ag + (rbase + r) * (size_t)KDIM + k0 + c4;
      float* lp = &sm.stg.a[buf_][r * A_STR + c4];
#if USE_ASYNC
      ASYNC_CP_B128(gp, lp);
#else
      *(float4*)lp = *(const float4*)gp;
#endif
    }
    #pragma unroll
    for (int i = 0; i < 16; ++i) {                // B tile: 32 K-rows x 128 cols
      const int idx = tid + i * 256;
      const int c   = idx & 127;
      const int kr  = idx >> 7;
      int col = col0_ + c;
      col = (col < MDIM) ? col : (MDIM - 1);      // clamp: uniform op count, pad
      const float* gp = Bg + (size_t)(k0 + kr) * MDIM + col;   // cols killed by cn=1e30
      float* lp = &sm.stg.b[buf_][(((kr >> 1) * CSTR + c) << 1) + (kr & 1)];
#if USE_ASYNC
      ASYNC_CP_B32(gp, lp);
#else
      *lp = *gp;
#endif
    }
  };

  // row norms for the 8 C/D rows this lane owns (C layout: M = v + 8*khalf)
  float rloc[8];
  #pragma unroll
  for (int v = 0; v < 8; ++v) rloc[v] = rn[rbase + wave * 16 + khalf * 8 + v];

  // running top-3 of squared distance per owned row
  float t0[8], t1[8], t2[8];
  #pragma unroll
  for (int v = 0; v < 8; ++v) { t0[v] = 3e38f; t1[v] = 3e38f; t2[v] = 3e38f; }

  const v8f vzero = {};

  for (int nt = 0; nt < NITER; ++nt) {
    const int col0 = nt * NTILE;

    v8f acc[8];
    #pragma unroll
    for (int t = 0; t < 8; ++t) acc[t] = vzero;

    stage(col0, 0, 0);                            // pipeline prologue

    for (int kt = 0; kt < KITER; ++kt) {
      const int cur = kt & 1;
      if (kt + 1 < KITER) {
        stage(col0, kt + 1, cur ^ 1);             // overlap next chunk's copies
        WAIT_ASYNC(20);                           // previous 20 done; new 20 in flight
      } else {
        WAIT_ASYNC(0);
      }
      if (kt + 2 < KITER) {                       // L2 prefetch two chunks ahead
        const int k2 = (kt + 2) * KT;
        __builtin_prefetch(Ag + (rbase + (tid >> 1)) * (size_t)KDIM + k2 + (tid & 1) * 16, 0, 3);
        __builtin_prefetch(Bg + (size_t)(k2 + (tid >> 3)) * MDIM + col0 + (tid & 7) * 16, 0, 3);
      }
      __syncthreads();                            // all waves' chunk-kt data ready

      // WMMA over this K chunk: 8 k-steps x 8 col tiles, pure b64 frag loads
      const float* Ab = &sm.stg.a[cur][(wave * 16 + lrow) * A_STR + (khalf << 1)];
      const float* Bb = &sm.stg.b[cur][(khalf * CSTR + lrow) << 1];
      for (int kk = 0; kk < KT; kk += 4) {
        const v2f a = *(const v2f*)(Ab + kk);     // A[16w+lrow][kk+2kh .. +1]
        #pragma unroll
        for (int t = 0; t < 8; ++t) {
          const v2f b = *(const v2f*)(Bb + (((kk >> 1) * CSTR + t * 16) << 1));
          acc[t] = __builtin_amdgcn_wmma_f32_16x16x4_f32(
              false, a, false, b, (short)0, acc[t], false, false);
        }
      }
      __syncthreads();                            // reads done before buffer reuse
    } // kt

    // epilogue: dist^2 and running top-3 insertion
    #pragma unroll
    for (int t = 0; t < 8; ++t) {
      const int col = col0 + t * 16 + lrow;       // < MPAD always
      const float cv = cn[col];
      #pragma unroll
      for (int v = 0; v < 8; ++v) {
        const float d2 = rloc[v] + cv - 2.0f * acc[t][v];
        if (d2 < t0[v])      { t2[v] = t1[v]; t1[v] = t0[v]; t0[v] = d2; }
        else if (d2 < t1[v]) { t2[v] = t1[v]; t1[v] = d2; }
        else if (d2 < t2[v]) { t2[v] = d2; }
      }
    }
  } // nt

  // cross-lane merge: each row's 16 partial top-3 sets -> final top-3
  __syncthreads();
  #pragma unroll
  for (int v = 0; v < 8; ++v) {
    const int r = wave * 16 + khalf * 8 + v;
    float* p = &sm.red[(r * 16 + lrow) * 3];
    p[0] = t0[v]; p[1] = t1[v]; p[2] = t2[v];
  }
  __syncthreads();
  if (tid < RT) {
    float b0 = 3e38f, b1 = 3e38f, b2 = 3e38f;
    const float* p = &sm.red[tid * 48];
    for (int j = 0; j < 48; ++j) {
      const float d = p[j];
      if (d < b0)      { b2 = b1; b1 = b0; b0 = d; }
      else if (d < b1) { b2 = b1; b1 = d; }
      else if (d < b2) { b2 = d; }
    }
    const float d0 = sqrtf(b0), d1 = sqrtf(b1), d2v = sqrtf(b2);
    // softmin over (d0,d1,d2), weight of smallest; stable since d0 <= d1,d2
    const float w0 = 1.0f / (1.0f + __expf(d0 - d1) + __expf(d0 - d2v));
    out[rbase + tid] = w0 * d0;
  }
}

// ---------------------------------------------------------------------------

extern "C" void kernel_launch(void* const* d_in, const int* in_sizes, int n_in,
                              void* d_out, int out_size, void* d_ws, size_t ws_size,
                              hipStream_t stream) {
  const float* phi = (const float*)d_in[0];   // [16, 3136, 1792] fp32
  const float* Cb  = (const float*)d_in[1];   // [1792, 3136] fp32
  float* out = (float*)d_out;                 // [16, 3136, 1] fp32

  float* rn = (float*)d_ws;                   // [50176]
  float* cn = rn + R_TOTAL;                   // [3200]

  row_norms_k<<<R_TOTAL / 8, 256, 0, stream>>>(phi, rn);
  col_norms_k<<<(MPAD + 255) / 256, 256, 0, stream>>>(Cb, cn);
  cfa_topk_k<<<R_TOTAL / RT, 256, 0, stream>>>(phi, Cb, rn, cn, out);
}